// GNNEmbeddingmix_5540507812307
// MI455X (gfx1250) — compile-verified
//
#include <hip/hip_runtime.h>
#include <hip/hip_bf16.h>

#define N_NODES   100000
#define N_EDGES   1000000
#define D_IN      64
#define EMB       6
#define D_OUT     64
#define N_TAIL    1000
#define D_NODE    70       // D_IN + EMB
#define D_EDGE_IN 204      // 3*D_IN + 2*EMB

typedef float v2f __attribute__((ext_vector_type(2)));
typedef float v8f __attribute__((ext_vector_type(8)));

// ---------------------------------------------------------------- zero init
__global__ void k_zero(float* __restrict__ node_out, float* __restrict__ den,
                       float* __restrict__ mean) {
  long long i = (long long)blockIdx.x * blockDim.x + threadIdx.x;
  const long long nnode = (long long)N_NODES * D_NODE;
  if (i < nnode)                    node_out[i] = 0.0f;
  else if (i < nnode + N_NODES)     den[i - nnode] = 0.0f;
  else if (i < nnode + N_NODES + D_NODE) mean[i - nnode - N_NODES] = 0.0f;
}

// ------------------------------------------------- masked scatter-sum (atomics)
__global__ void k_scatter(const float* __restrict__ edge_attr,
                          const float* __restrict__ mask,
                          const int* __restrict__ col,
                          float* __restrict__ node_out,
                          float* __restrict__ den) {
  long long i = (long long)blockIdx.x * blockDim.x + threadIdx.x;
  if (i >= (long long)N_EDGES * D_IN) return;
  int e = (int)(i >> 6);
  int f = (int)(i & 63);
  float m = mask[e];
  if (m != 0.0f) {
    int c = col[e];
    atomicAdd(&node_out[(long long)c * D_NODE + f], edge_attr[i] * m);
    if (f == 0) atomicAdd(&den[c], m);
  }
}

// ------------------------------------------- normalize + append x features
__global__ void k_norm(const float* __restrict__ x, const float* __restrict__ den,
                       float* __restrict__ node_out) {
  long long i = (long long)blockIdx.x * blockDim.x + threadIdx.x;
  if (i >= (long long)N_NODES * D_NODE) return;
  int n = (int)(i / D_NODE);
  int f = (int)(i - (long long)n * D_NODE);
  if (f < D_IN) node_out[i] = node_out[i] / (den[n] + 1.0f);
  else          node_out[i] = x[(long long)n * EMB + (f - D_IN)];
}

// -------------------------------- tail mix, pass 1: gather snapshot (+mean)
__global__ void k_tail_gather(const int* __restrict__ change,
                              const int* __restrict__ is_support,
                              const int* __restrict__ tail,
                              const float* __restrict__ support_tail,
                              const float* __restrict__ node_out,
                              float* __restrict__ buf, float* __restrict__ mean) {
  int i = blockIdx.x * blockDim.x + threadIdx.x;
  if (i >= N_TAIL * D_NODE) return;
  int t = i / D_NODE, f = i - t * D_NODE;
  int n = tail[t];
  float v = node_out[(long long)n * D_NODE + f];
  int ch = change[0], sup = is_support[0];
  if (ch && sup)      atomicAdd(&mean[f], v * (1.0f / (float)N_TAIL));
  else if (ch)        v = 0.1f * support_tail[f] + 0.9f * v;
  buf[i] = v;
}

// -------------------------------- tail mix, pass 2: scatter (dup-safe)
__global__ void k_tail_scatter(const int* __restrict__ change,
                               const int* __restrict__ is_support,
                               const int* __restrict__ tail,
                               const float* __restrict__ buf,
                               const float* __restrict__ mean,
                               float* __restrict__ node_out) {
  int i = blockIdx.x * blockDim.x + threadIdx.x;
  if (i >= N_TAIL * D_NODE) return;
  int t = i / D_NODE, f = i - t * D_NODE;
  int n = tail[t];
  float v = (change[0] && is_support[0]) ? mean[f] : buf[i];
  node_out[(long long)n * D_NODE + f] = v;
}

// ---------------------------------------------------------------- edge GEMM
// Per block: 8 waves, each wave computes 16 edges x 64 outputs with
// v_wmma_f32_16x16x4_f32 (fp32-exact path, K=204 in 51 steps of 4).
// LDS: W transposed [64][210] + bias [64] + per-wave A [16][210] + indices.
// Stride 210: even (8B-aligned b64 frags at even k), 210%64=18 -> no bank
// conflicts across a 16-lane group.
#define TE_WAVE  16
#define TE_BLOCK 128
#define A_STRIDE 210
#define W_STRIDE 210
#define LDS_WT_FLOATS  (D_OUT * W_STRIDE)            // 13440
#define LDS_BIAS_OFF   LDS_WT_FLOATS                 // 13440
#define LDS_A_OFF      (LDS_WT_FLOATS + D_OUT)       // 13504
#define LDS_A_FLOATS   (8 * TE_WAVE * A_STRIDE)      // 26880
#define LDS_IDX_OFF    (LDS_A_OFF + LDS_A_FLOATS)    // 40384
#define LDS_TOTAL_F    (LDS_IDX_OFF + 8 * TE_WAVE * 2) // 40640 floats = 162560 B

__global__ void k_edge_gemm(const float* __restrict__ node_rep,
                            const float* __restrict__ edge_attr,
                            const int* __restrict__ eidx,    // [2][E]
                            const float* __restrict__ W,     // [204][64]
                            const float* __restrict__ bias,  // [64]
                            float* __restrict__ edge_out) {  // [E][64]
  extern __shared__ float smem[];
  float* sWt   = smem;                       // transposed: sWt[n*210 + k]
  float* sBias = smem + LDS_BIAS_OFF;
  float* sA    = smem + LDS_A_OFF;
  int*   sIdx  = (int*)(smem + LDS_IDX_OFF);

  const int tid  = threadIdx.x;
  const int wave = tid >> 5;
  const int lane = tid & 31;
  // e0 is uniform across the wave; make that visible to the compiler so the
  // tile-bounds tests become scalar branches (s_cmp/s_cbranch, no exec dance).
  const int e0 = __builtin_amdgcn_readfirstlane(blockIdx.x * TE_BLOCK + wave * TE_WAVE);
  const bool fullTile = (e0 + TE_WAVE <= N_EDGES);

  // stage W transposed (coalesced global read, strided LDS write) + bias
  for (int i = tid; i < D_EDGE_IN * D_OUT; i += 256) {
    int k = i >> 6;         // 0..203
    int n = i & 63;         // 0..63
    sWt[n * W_STRIDE + k] = W[i];
  }
  if (tid < D_OUT) sBias[tid] = bias[tid];

  // stage this wave's 16 edge index pairs
  if (lane < TE_WAVE) {
    int e = e0 + lane; if (e >= N_EDGES) e = N_EDGES - 1;
    sIdx[(wave * TE_WAVE + lane) * 2 + 0] = eidx[e];            // row
    sIdx[(wave * TE_WAVE + lane) * 2 + 1] = eidx[N_EDGES + e];  // col
  }
  __syncthreads();

  // stage A rows: concat(node_rep[row], node_rep[col], edge_attr[e]) = 204 floats
  float* sAw = sA + wave * TE_WAVE * A_STRIDE;
  for (int i = lane; i < TE_WAVE * D_EDGE_IN; i += 32) {
    int el = i / D_EDGE_IN;
    int k  = i - el * D_EDGE_IN;
    int e  = e0 + el; if (e >= N_EDGES) e = N_EDGES - 1;
    int r  = sIdx[(wave * TE_WAVE + el) * 2 + 0];
    int c  = sIdx[(wave * TE_WAVE + el) * 2 + 1];
    float v;
    if (k < D_NODE)          v = node_rep[(long long)r * D_NODE + k];
    else if (k < 2 * D_NODE) v = node_rep[(long long)c * D_NODE + (k - D_NODE)];
    else                     v = edge_attr[(long long)e * D_IN + (k - 2 * D_NODE)];
    sAw[el * A_STRIDE + k] = v;
  }
  __syncthreads();

  const int halfLane = lane & 15;
  const int koff     = (lane < 16) ? 0 : 2;  // A/B frag layout: lane halves hold K, K+2

  // accumulators: 4 N-tiles of 16; init with bias (D layout: N = t*16 + halfLane)
  v8f acc0, acc1, acc2, acc3;
  {
    float b0 = sBias[ 0 + halfLane];
    float b1 = sBias[16 + halfLane];
    float b2 = sBias[32 + halfLane];
    float b3 = sBias[48 + halfLane];
    for (int j = 0; j < 8; ++j) { acc0[j] = b0; acc1[j] = b1; acc2[j] = b2; acc3[j] = b3; }
  }

  // all fragment loads are 8B-aligned contiguous pairs -> single ds_load_b64 each
  const float* aRow = sAw + halfLane * A_STRIDE + koff;
  const float* w0   = sWt + ( 0 + halfLane) * W_STRIDE + koff;
  const float* w1   = sWt + (16 + halfLane) * W_STRIDE + koff;
  const float* w2   = sWt + (32 + halfLane) * W_STRIDE + koff;
  const float* w3   = sWt + (48 + halfLane) * W_STRIDE + koff;

  for (int k0 = 0; k0 < D_EDGE_IN; k0 += 4) {
    v2f a = *(const v2f*)(aRow + k0);
    acc0 = __builtin_amdgcn_wmma_f32_16x16x4_f32(false, a, false, *(const v2f*)(w0 + k0),
                                                 (short)0, acc0, false, false);
    acc1 = __builtin_amdgcn_wmma_f32_16x16x4_f32(false, a, false, *(const v2f*)(w1 + k0),
                                                 (short)0, acc1, false, false);
    acc2 = __builtin_amdgcn_wmma_f32_16x16x4_f32(false, a, false, *(const v2f*)(w2 + k0),
                                                 (short)0, acc2, false, false);
    acc3 = __builtin_amdgcn_wmma_f32_16x16x4_f32(false, a, false, *(const v2f*)(w3 + k0),
                                                 (short)0, acc3, false, false);
  }

  // store D: VGPR j -> edge e0 + j (+8 for upper lane half); feature t*16 + halfLane
  const int mbase = (lane < 16) ? 0 : 8;
  if (fullTile) {
    // full tile (scalar branch): one base pointer, all offsets immediate
    float* outp = edge_out + (long long)(e0 + mbase) * D_OUT + halfLane;
#pragma unroll
    for (int j = 0; j < 8; ++j) {
      outp[j * D_OUT +  0] = acc0[j];
      outp[j * D_OUT + 16] = acc1[j];
      outp[j * D_OUT + 32] = acc2[j];
      outp[j * D_OUT + 48] = acc3[j];
    }
  } else {
#pragma unroll
    for (int j = 0; j < 8; ++j) {
      int e = e0 + mbase + j;
      if (e < N_EDGES) {
        long long base = (long long)e * D_OUT + halfLane;
        edge_out[base +  0] = acc0[j];
        edge_out[base + 16] = acc1[j];
        edge_out[base + 32] = acc2[j];
        edge_out[base + 48] = acc3[j];
      }
    }
  }
}

// ---------------------------------------------------------------- launcher
extern "C" void kernel_launch(void* const* d_in, const int* in_sizes, int n_in,
                              void* d_out, int out_size, void* d_ws, size_t ws_size,
                              hipStream_t stream) {
  const int*   change       = (const int*)d_in[0];
  const int*   is_support   = (const int*)d_in[1];
  // d_in[2] = head_idxs (unused by reference)
  const int*   tail_idxs    = (const int*)d_in[3];
  const float* x            = (const float*)d_in[4];
  const int*   edge_index   = (const int*)d_in[5];   // [2][E]
  const float* edge_attr    = (const float*)d_in[6]; // [E][64]
  const float* mask         = (const float*)d_in[7];
  // d_in[8] = num_nodes (fixed)
  const float* support_tail = (const float*)d_in[9];
  const float* W_edge       = (const float*)d_in[10]; // [204][64]
  const float* b_edge       = (const float*)d_in[11]; // [64]

  float* node_out = (float*)d_out;                                  // [N][70]
  float* edge_out = node_out + (long long)N_NODES * D_NODE;         // [E][64]

  float* den  = (float*)d_ws;            // N floats
  float* buf  = den + N_NODES;           // N_TAIL * 70 floats
  float* mean = buf + N_TAIL * D_NODE;   // 70 floats

  {
    long long tot = (long long)N_NODES * D_NODE + N_NODES + D_NODE;
    k_zero<<<(int)((tot + 255) / 256), 256, 0, stream>>>(node_out, den, mean);
  }
  {
    long long tot = (long long)N_EDGES * D_IN;
    k_scatter<<<(int)((tot + 255) / 256), 256, 0, stream>>>(
        edge_attr, mask, edge_index + N_EDGES, node_out, den);
  }
  {
    long long tot = (long long)N_NODES * D_NODE;
    k_norm<<<(int)((tot + 255) / 256), 256, 0, stream>>>(x, den, node_out);
  }
  {
    int tot = N_TAIL * D_NODE;
    k_tail_gather<<<(tot + 255) / 256, 256, 0, stream>>>(
        change, is_support, tail_idxs, support_tail, node_out, buf, mean);
    k_tail_scatter<<<(tot + 255) / 256, 256, 0, stream>>>(
        change, is_support, tail_idxs, buf, mean, node_out);
  }
  {
    int blocks = (N_EDGES + TE_BLOCK - 1) / TE_BLOCK;
    size_t lds = (size_t)LDS_TOTAL_F * sizeof(float);  // 162560 B (<= 320KB WGP LDS)
    k_edge_gemm<<<blocks, 256, lds, stream>>>(
        node_out, edge_attr, edge_index, W_edge, b_edge, edge_out);
  }
}